// EntityEncoder_23776938951190
// MI455X (gfx1250) — compile-verified
//
#include <hip/hip_runtime.h>
#include <hip/hip_bf16.h>

// ---------------------------------------------------------------------------
// EntityEncoder on MI455X (gfx1250): fp32 transformer encoder built on
// V_WMMA_F32_16X16X4_F32 (native fp32 matrix op, wave32).
// Round 2: A-fragment register reuse (QKV share A; attention hoists Q tile;
// FFN1 shares A across both column tiles) to cut VMEM-per-WMMA.
// ---------------------------------------------------------------------------

typedef __attribute__((ext_vector_type(2))) float v2f;
typedef __attribute__((ext_vector_type(8))) float v8f;

#define A_DIM 256
#define N_SEQ 1024
#define B_SZ  16
#define H_HEADS 2
#define D_HEAD 128
#define FFN_DIM 1024

__device__ __forceinline__ v8f wmma4(v2f a, v2f b, v8f c) {
  // D = A(16x4,f32) * B(4x16,f32) + C(16x16,f32)
  return __builtin_amdgcn_wmma_f32_16x16x4_f32(
      /*neg_a=*/false, a, /*neg_b=*/false, b,
      /*c_mod=*/(short)0, c, /*reuse_a=*/false, /*reuse_b=*/false);
}

// A-fragment: 16x4 tile at (row0,k0) of row-major [*, ld].
// lanes 0-15 hold K={k0,k0+1} of row row0+lane; lanes 16-31 hold K={k0+2,k0+3}.
__device__ __forceinline__ v2f load_afrag(const float* __restrict__ p, int ld,
                                          int row0, int k0) {
  const int l = threadIdx.x & 31;
  const int row = row0 + (l & 15);
  const int kk = k0 + 2 * (l >> 4);
  const float* q = p + (long)row * ld + kk;
  v2f a; a[0] = q[0]; a[1] = q[1]; return a;
}

// B-fragment: 4x16 tile at (k0,col0) of row-major [*, ld].
// lanes 0-15 hold rows K={k0,k0+1} at col col0+lane; lanes 16-31 rows {k0+2,k0+3}.
__device__ __forceinline__ v2f load_bfrag(const float* __restrict__ p, int ld,
                                          int k0, int col0) {
  const int l = threadIdx.x & 31;
  const int col = col0 + (l & 15);
  const int kk = k0 + 2 * (l >> 4);
  v2f b;
  b[0] = p[(long)kk * ld + col];
  b[1] = p[(long)(kk + 1) * ld + col];
  return b;
}

// ---------------------------------------------------------------------------
// QKV projection: one A-fragment load feeds three WMMAs (Q, K, V).
// grid = M/16 blocks, 512 threads (16 waves); wave w owns output cols w*16..
// ---------------------------------------------------------------------------
__global__ __launch_bounds__(512) void k_proj_qkv(
    const float* __restrict__ Xin,
    const float* __restrict__ Wq, const float* __restrict__ bq,
    const float* __restrict__ Wk, const float* __restrict__ bk,
    const float* __restrict__ Wv, const float* __restrict__ bv,
    float* __restrict__ Q, float* __restrict__ K, float* __restrict__ V) {
  const int m0 = blockIdx.x * 16;
  const int wid = threadIdx.x >> 5;
  const int n0 = wid * 16;
  const int l = threadIdx.x & 31;
  const int half = l >> 4, lc = l & 15;

  v8f accq = {}, acck = {}, accv = {};
  for (int k0 = 0; k0 < A_DIM; k0 += 4) {
    const v2f a = load_afrag(Xin, A_DIM, m0, k0);
    accq = wmma4(a, load_bfrag(Wq, A_DIM, k0, n0), accq);
    acck = wmma4(a, load_bfrag(Wk, A_DIM, k0, n0), acck);
    accv = wmma4(a, load_bfrag(Wv, A_DIM, k0, n0), accv);
  }
  const float biq = bq[n0 + lc];
  const float bik = bk[n0 + lc];
  const float biv = bv[n0 + lc];
  for (int j = 0; j < 8; ++j) {
    const long row = m0 + j + 8 * half;
    Q[row * A_DIM + n0 + lc] = accq[j] + biq;
    K[row * A_DIM + n0 + lc] = acck[j] + bik;
    V[row * A_DIM + n0 + lc] = accv[j] + biv;
  }
}

// ---------------------------------------------------------------------------
// Flash-style attention. One block = (b, h, 16 query rows); 4 waves.
// Q tile (16x128 = 32 fragments, 64 VGPRs) hoisted into registers; per KV
// tile each wave computes S via WMMA, does online softmax with 16-lane
// shuffles, bounces P through LDS into A-fragment layout, accumulates P@V.
// ---------------------------------------------------------------------------
__global__ __launch_bounds__(128) void k_attn(
    const float* __restrict__ Q, const float* __restrict__ K,
    const float* __restrict__ V, float* __restrict__ Oout) {
  __shared__ float ldsP[4][16 * 16];
  const int wid = threadIdx.x >> 5;
  const int l = threadIdx.x & 31;
  const int half = l >> 4, lc = l & 15;

  const int qt = blockIdx.x & 63;  // N/16 = 64 query tiles
  const int bh = blockIdx.x >> 6;  // 0..31
  const int h = bh & (H_HEADS - 1);
  const int b = bh >> 1;
  const long base = (long)b * N_SEQ * A_DIM;
  const int hoff = h * D_HEAD;
  const int m0 = qt * 16;
  const float scale = 0.088388347648318447f;  // 1/sqrt(128)

  const float* Qp = Q + base + hoff;
  const float* Kp = K + base + hoff;
  const float* Vp = V + base + hoff;

  // Hoist the wave's Q tile: invariant across all 64 KV tiles.
  v2f qa[D_HEAD / 4];
#pragma unroll
  for (int k4 = 0; k4 < D_HEAD / 4; ++k4) {
    const int kk = 4 * k4 + 2 * half;
    const float* p = Qp + (long)(m0 + lc) * A_DIM + kk;
    qa[k4][0] = p[0];
    qa[k4][1] = p[1];
  }

  float mrun[8], lrun[8];
  v8f o0 = {}, o1 = {};
  for (int j = 0; j < 8; ++j) { mrun[j] = -1e30f; lrun[j] = 0.0f; }

  for (int kt = 0; kt < N_SEQ / 16; ++kt) {
    // ---- S = Q * K^T (16x16 tile), K=128: only B fragments from memory.
    v8f s = {};
#pragma unroll
    for (int k4 = 0; k4 < D_HEAD / 4; ++k4) {
      v2f bf;  // B[d][n] = K[kt*16+n][d]: row-major K gives the transpose free
      const int kk = 4 * k4 + 2 * half;
      const float* p = Kp + (long)(kt * 16 + lc) * A_DIM + kk;
      bf[0] = p[0]; bf[1] = p[1];
      s = wmma4(qa[k4], bf, s);
    }
    // ---- online softmax (rows live across 16-lane groups in C layout)
    float alpha[8], prob[8];
    for (int j = 0; j < 8; ++j) {
      float sv = s[j] * scale;
      float tmax = sv;
      for (int m = 1; m < 16; m <<= 1) tmax = fmaxf(tmax, __shfl_xor(tmax, m, 32));
      const float mnew = fmaxf(mrun[j], tmax);
      const float al = __expf(mrun[j] - mnew);
      const float p = __expf(sv - mnew);
      float rs = p;
      for (int m = 1; m < 16; m <<= 1) rs += __shfl_xor(rs, m, 32);
      lrun[j] = lrun[j] * al + rs;
      mrun[j] = mnew;
      alpha[j] = al; prob[j] = p;
    }
    for (int j = 0; j < 8; ++j) { o0[j] *= alpha[j]; o1[j] *= alpha[j]; }
    // ---- P tile through LDS: C layout -> A-fragment layout
    __syncthreads();
    for (int j = 0; j < 8; ++j) ldsP[wid][(j + 8 * half) * 16 + lc] = prob[j];
    __syncthreads();
    // ---- O += P @ V (two 16-col tiles per wave), K=16 -> 4 WMMAs each
#pragma unroll
    for (int k4 = 0; k4 < 4; ++k4) {
      const int kk = 4 * k4 + 2 * half;
      v2f a; a[0] = ldsP[wid][lc * 16 + kk]; a[1] = ldsP[wid][lc * 16 + kk + 1];
      const float* p0 = Vp + (long)(kt * 16 + kk) * A_DIM + wid * 32 + lc;
      v2f bf0, bf1;
      bf0[0] = p0[0];      bf0[1] = p0[A_DIM];
      bf1[0] = p0[16];     bf1[1] = p0[A_DIM + 16];
      o0 = wmma4(a, bf0, o0);
      o1 = wmma4(a, bf1, o1);
    }
  }
  // ---- normalize and store
  for (int t = 0; t < 2; ++t) {
    const int n0w = wid * 32 + t * 16;
    const v8f o = t ? o1 : o0;
    for (int j = 0; j < 8; ++j) {
      const int row = m0 + j + 8 * half;
      Oout[base + (long)row * A_DIM + hoff + n0w + lc] = o[j] / lrun[j];
    }
  }
}

// ---------------------------------------------------------------------------
// Fused FFN (256->1024 ReLU ->256 ReLU) + residual + LayerNorm.
// Hidden kept in LDS in two 512-wide phases (32 KB), FFN2 accumulated
// across phases; LN done per-row by one wave with shuffle reductions.
// FFN1 shares each A fragment across the wave's two column tiles.
// ---------------------------------------------------------------------------
__global__ __launch_bounds__(512) void k_ffn_ln(
    const float* __restrict__ Attn,
    const float* __restrict__ W1, const float* __restrict__ b1,
    const float* __restrict__ W2, const float* __restrict__ b2,
    const float* __restrict__ Step_in,
    const float* __restrict__ ln_g, const float* __restrict__ ln_b,
    float* __restrict__ Step_out) {
  __shared__ float hid[16 * 512];  // also reused as 16x256 LN buffer
  const int m0 = blockIdx.x * 16;
  const int wid = threadIdx.x >> 5;
  const int l = threadIdx.x & 31;
  const int half = l >> 4, lc = l & 15;

  v8f acc2 = {};
  for (int ph = 0; ph < 2; ++ph) {
    const int hb = ph * 512;
    // FFN1: 512 hidden cols -> 32 tiles / 16 waves = 2 tiles each. K=256.
    const int n0a = wid * 32 + hb;
    const int n0b = n0a + 16;
    v8f h0 = {}, h1 = {};
    for (int k0 = 0; k0 < A_DIM; k0 += 4) {
      const v2f a = load_afrag(Attn, A_DIM, m0, k0);
      h0 = wmma4(a, load_bfrag(W1, FFN_DIM, k0, n0a), h0);
      h1 = wmma4(a, load_bfrag(W1, FFN_DIM, k0, n0b), h1);
    }
    const float bi0 = b1[n0a + lc];
    const float bi1 = b1[n0b + lc];
    __syncthreads();  // previous-phase readers done
    for (int j = 0; j < 8; ++j) {
      const int row = j + 8 * half;
      hid[row * 512 + wid * 32 + lc]      = fmaxf(h0[j] + bi0, 0.0f);
      hid[row * 512 + wid * 32 + 16 + lc] = fmaxf(h1[j] + bi1, 0.0f);
    }
    __syncthreads();
    // FFN2 partial: wave owns output col tile wid*16; K = 512 this phase.
    const int n0 = wid * 16;
    const float* W2p = W2 + (long)hb * A_DIM;
    for (int k0 = 0; k0 < 512; k0 += 4) {
      v2f a;
      {
        const int kk = k0 + 2 * half;
        a[0] = hid[lc * 512 + kk];
        a[1] = hid[lc * 512 + kk + 1];
      }
      acc2 = wmma4(a, load_bfrag(W2p, A_DIM, k0, n0), acc2);
    }
    __syncthreads();
  }
  // Epilogue: bias + ReLU + residual -> LDS for LayerNorm.
  const int n0 = wid * 16;
  const float bias = b2[n0 + lc];
  for (int j = 0; j < 8; ++j) {
    const int row = j + 8 * half;
    const float y = fmaxf(acc2[j] + bias, 0.0f) +
                    Step_in[(long)(m0 + row) * A_DIM + n0 + lc];
    hid[row * A_DIM + n0 + lc] = y;
  }
  __syncthreads();
  // LayerNorm: wave r normalizes row r (16 waves, 16 rows; 8 elems/lane).
  {
    const int r = wid;
    float vals[8], vsum = 0.0f, vsq = 0.0f;
    for (int c8 = 0; c8 < 8; ++c8) {
      const float v = hid[r * A_DIM + c8 * 32 + l];
      vals[c8] = v; vsum += v; vsq += v * v;
    }
    for (int m = 1; m < 32; m <<= 1) {
      vsum += __shfl_xor(vsum, m, 32);
      vsq += __shfl_xor(vsq, m, 32);
    }
    const float mu = vsum * (1.0f / A_DIM);
    const float var = vsq * (1.0f / A_DIM) - mu * mu;
    const float rinv = rsqrtf(var + 1e-5f);
    for (int c8 = 0; c8 < 8; ++c8) {
      const int c = c8 * 32 + l;
      Step_out[(long)(m0 + r) * A_DIM + c] =
          (vals[c8] - mu) * rinv * ln_g[c] + ln_b[c];
    }
  }
}

// ---------------------------------------------------------------------------
// Conv1d(k=1) + ReLU == GEMM with ReLU epilogue.
// ---------------------------------------------------------------------------
__global__ __launch_bounds__(512) void k_conv(
    const float* __restrict__ S, const float* __restrict__ Wc,
    const float* __restrict__ bc, float* __restrict__ E) {
  const int m0 = blockIdx.x * 16;
  const int wid = threadIdx.x >> 5;
  const int l = threadIdx.x & 31;
  const int half = l >> 4, lc = l & 15;
  const int n0 = wid * 16;
  v8f acc = {};
  for (int k0 = 0; k0 < A_DIM; k0 += 4) {
    const v2f a = load_afrag(S, A_DIM, m0, k0);
    acc = wmma4(a, load_bfrag(Wc, A_DIM, k0, n0), acc);
  }
  const float bias = bc[n0 + lc];
  for (int j = 0; j < 8; ++j) {
    const int row = m0 + j + 8 * half;
    E[(long)row * A_DIM + n0 + lc] = fmaxf(acc[j] + bias, 0.0f);
  }
}

// Mean pool over sequence: pooled[b][c] = mean_n E[b,n,c].
__global__ void k_pool(const float* __restrict__ E, float* __restrict__ P) {
  const int b = blockIdx.x;
  const int c = threadIdx.x;
  const float* p = E + (long)b * N_SEQ * A_DIM + c;
  float s = 0.0f;
  for (int n = 0; n < N_SEQ; ++n) s += p[(long)n * A_DIM];
  P[b * A_DIM + c] = s * (1.0f / N_SEQ);
}

// Final head: relu(pooled @ We + be). Tiny (2 MFLOP): plain VALU dot.
__global__ void k_final(const float* __restrict__ P, const float* __restrict__ We,
                        const float* __restrict__ be, float* __restrict__ Out) {
  const int b = blockIdx.x;
  const int c = threadIdx.x;
  float s = be[c];
  const float* pr = P + b * A_DIM;
  for (int k = 0; k < A_DIM; ++k) s += pr[k] * We[k * A_DIM + c];
  Out[b * A_DIM + c] = fmaxf(s, 0.0f);
}

// ---------------------------------------------------------------------------
extern "C" void kernel_launch(void* const* d_in, const int* in_sizes, int n_in,
                              void* d_out, int out_size, void* d_ws, size_t ws_size,
                              hipStream_t stream) {
  (void)in_sizes; (void)n_in; (void)out_size; (void)ws_size;
  const float* X   = (const float*)d_in[0];
  const float* Wq  = (const float*)d_in[1];
  const float* bq  = (const float*)d_in[2];
  const float* Wk  = (const float*)d_in[3];
  const float* bk  = (const float*)d_in[4];
  const float* Wv  = (const float*)d_in[5];
  const float* bv  = (const float*)d_in[6];
  const float* W1  = (const float*)d_in[7];
  const float* b1  = (const float*)d_in[8];
  const float* W2  = (const float*)d_in[9];
  const float* b2  = (const float*)d_in[10];
  const float* lng = (const float*)d_in[11];
  const float* lnb = (const float*)d_in[12];
  const float* Wc  = (const float*)d_in[13];
  const float* bc  = (const float*)d_in[14];
  const float* We  = (const float*)d_in[15];
  const float* be  = (const float*)d_in[16];

  const long ACT = (long)B_SZ * N_SEQ * A_DIM;  // 4,194,304 floats
  float* ws = (float*)d_ws;
  float* Qb = ws + 0 * ACT;
  float* Kb = ws + 1 * ACT;
  float* Vb = ws + 2 * ACT;
  float* Ab = ws + 3 * ACT;
  float* Sb = ws + 4 * ACT;   // running step
  float* Pb = ws + 5 * ACT;   // pooled [16,256]
  float* Eb = Qb;             // entity encodings reuse Q buffer

  const int MB = (B_SZ * N_SEQ) / 16;  // 1024 row-tile blocks
  const float* step = X;
  for (int i = 0; i < 3; ++i) {
    k_proj_qkv<<<MB, 512, 0, stream>>>(
        step,
        Wq + (long)i * A_DIM * A_DIM, bq + i * A_DIM,
        Wk + (long)i * A_DIM * A_DIM, bk + i * A_DIM,
        Wv + (long)i * A_DIM * A_DIM, bv + i * A_DIM,
        Qb, Kb, Vb);
    k_attn<<<B_SZ * H_HEADS * (N_SEQ / 16), 128, 0, stream>>>(Qb, Kb, Vb, Ab);
    k_ffn_ln<<<MB, 512, 0, stream>>>(
        Ab,
        W1 + (long)i * A_DIM * FFN_DIM, b1 + i * FFN_DIM,
        W2 + (long)i * FFN_DIM * A_DIM, b2 + i * A_DIM,
        step, lng, lnb, Sb);
    step = Sb;
  }
  k_conv<<<MB, 512, 0, stream>>>(step, Wc, bc, Eb);
  k_pool<<<B_SZ, A_DIM, 0, stream>>>(Eb, Pb);
  k_final<<<B_SZ, A_DIM, 0, stream>>>(Pb, We, be, (float*)d_out);
}